// BFGSMat_1571958030813
// MI455X (gfx1250) — compile-verified
//
#include <hip/hip_runtime.h>

// L-BFGS apply_Hv via the compact (Gram-matrix) reformulation.
//  Phase 0: zpad_init                   -> 64-float zero region for pad-row loads
//  Phase 1: gram_partial + gram_reduce  -> S.Y^T, Y.Y^T, S.v, Y.v via V_WMMA_F32_16X16X4_F32
//  Phase 2: solve (1 thread)            -> alpha/beta triangular recurrences, emits 61 coeffs
//  Phase 3: combine                     -> out = (a/th) v - sum alpha_k/th y_k + sum (alpha_k-beta_k) s_k
// Total HBM traffic ~1.96 GB (== minimum for the two-loop recursion) at 23.3 TB/s -> ~84 us floor.

typedef float v2f __attribute__((ext_vector_type(2)));
typedef float v4f __attribute__((ext_vector_type(4)));
typedef float v8f __attribute__((ext_vector_type(8)));

// explicit global-address-space pointer to a primitive ext-vector: forces
// global_load_b128 (LOADcnt only), not flat_load_b128 (which also ties up
// DScnt and does per-lane aperture checks). ext-vector (not HIP float4)
// because HIP_vector_type's member operator= can't bind an AS(1) reference.
typedef const __attribute__((address_space(1))) v4f* gptr4;

#define MHIST       30
#define BK          64          // columns of Z staged per LDS tile
#define LDS_STRIDE  68          // 64 + 4 pad: fragment ds_load_b64 hits all 64 banks
#define NBLK1       256         // blocks for the gram pass
#define GRAM_ELEMS  2048        // 8 tiles * 256 elems
#define COEFF_OFF   2048        // float offset in ws: [cv, cy[30], cs[30]]
#define ZPAD_OFF    3072        // float offset in ws: 64 zeros for pad-row dummy loads
#define PART_OFF    4096        // float offset in ws: per-block partials

// Logical Z rows: 0..29 = S, 30..31 = 0, 32..61 = Y, 62 = v, 63 = 0.
// Gram tiles kept: i in 0..3 (all row tiles) x j in {2,3} (Y & v column tiles),
// stored as tileIdx = i*2 + (j-2), element (m,n) at tileIdx*256 + m*16 + n.

__global__ void zpad_init_kernel(float* __restrict__ ws)
{
  ws[ZPAD_OFF + threadIdx.x] = 0.f;
}

__global__ __launch_bounds__(256) void gram_partial_kernel(
    const float* __restrict__ v, const float* __restrict__ s,
    const float* __restrict__ y, float* __restrict__ ws, long n)
{
  __shared__ float lds[64 * LDS_STRIDE];   // 17.4 KB, reused for gram reduction

  const int tid  = threadIdx.x;
  const int wave = tid >> 5;
  const int lane = tid & 31;
  const int rsel = lane & 15;              // row within 16-row tile
  const int kh   = (lane >> 4) << 1;       // K half: lanes 0-15 -> K{0,1}, 16-31 -> K{2,3}

  // ---- loop-invariant row classification: walking pointers, zero-mask-free ----
  // Each thread owns 4 fixed rows (rr*16 + tid/16) x 4 fixed columns ((tid%16)*4).
  // Pad rows point at a 64-float zero region in ws and never advance, so staged
  // registers are raw load results (no dependent VALU -> loadcnt wait can sink
  // past the WMMA block, overlapping prefetch with compute).
  const int cg = (tid & 15) << 2;
  const long gstride = (long)gridDim.x * BK;
  const float* rp[4];
  long step[4];
  #pragma unroll
  for (int rr = 0; rr < 4; ++rr) {
    const int row = rr * 16 + (tid >> 4);
    const float* p; long st = gstride;
    if (row < MHIST)            p = s + (size_t)row * n + (size_t)blockIdx.x * BK;
    else if (row < 32)          { p = ws + ZPAD_OFF; st = 0; }     // pad -> zeros
    else if (row < 32 + MHIST)  p = y + (size_t)(row - 32) * n + (size_t)blockIdx.x * BK;
    else if (row == 62)         p = v + (size_t)blockIdx.x * BK;
    else                        { p = ws + ZPAD_OFF; st = 0; }     // pad -> zeros
    rp[rr]   = p + cg;
    step[rr] = st;
  }

  v8f C[8] = {};                           // 8 Gram tile accumulators

  long base = (long)blockIdx.x * BK;

  // register-staged prefetch of the first tile
  v4f stage[4];
  if (base < n) {
    #pragma unroll
    for (int rr = 0; rr < 4; ++rr) {
      stage[rr] = *(gptr4)(const void*)rp[rr];
      rp[rr] += step[rr];
    }
  }

  for (; base < n; base += gstride) {
    // ---- commit staged tile into LDS ----
    #pragma unroll
    for (int rr = 0; rr < 4; ++rr) {
      float* p = &lds[(rr * 16 + (tid >> 4)) * LDS_STRIDE + cg];
      p[0] = stage[rr].x; p[1] = stage[rr].y; p[2] = stage[rr].z; p[3] = stage[rr].w;
    }
    __syncthreads();

    // ---- prefetch next tile into registers (overlaps with wmma below) ----
    if (base + gstride < n) {              // uniform scalar branch
      #pragma unroll
      for (int rr = 0; rr < 4; ++rr) {
        stage[rr] = *(gptr4)(const void*)rp[rr];
        rp[rr] += step[rr];
      }
    }

    // ---- each wave: 2 K-chunks of 4 columns, 8 wmma each ----
    #pragma unroll
    for (int u = 0; u < 2; ++u) {
      const int c = (2 * wave + u) * 4;
      v2f f[4];
      #pragma unroll
      for (int t = 0; t < 4; ++t) {
        const float* p = &lds[(t * 16 + rsel) * LDS_STRIDE + c + kh];
        f[t].x = p[0]; f[t].y = p[1];      // A and B fragments share this layout for Gram
      }
      C[0] = __builtin_amdgcn_wmma_f32_16x16x4_f32(false, f[0], false, f[2], (short)0, C[0], false, false);
      C[1] = __builtin_amdgcn_wmma_f32_16x16x4_f32(false, f[0], false, f[3], (short)0, C[1], false, false);
      C[2] = __builtin_amdgcn_wmma_f32_16x16x4_f32(false, f[1], false, f[2], (short)0, C[2], false, false);
      C[3] = __builtin_amdgcn_wmma_f32_16x16x4_f32(false, f[1], false, f[3], (short)0, C[3], false, false);
      C[4] = __builtin_amdgcn_wmma_f32_16x16x4_f32(false, f[2], false, f[2], (short)0, C[4], false, false);
      C[5] = __builtin_amdgcn_wmma_f32_16x16x4_f32(false, f[2], false, f[3], (short)0, C[5], false, false);
      C[6] = __builtin_amdgcn_wmma_f32_16x16x4_f32(false, f[3], false, f[2], (short)0, C[6], false, false);
      C[7] = __builtin_amdgcn_wmma_f32_16x16x4_f32(false, f[3], false, f[3], (short)0, C[7], false, false);
    }
    __syncthreads();
  }

  // ---- deterministic wave-ordered reduction of C into LDS gram ----
  for (int e = tid; e < GRAM_ELEMS; e += 256) lds[e] = 0.f;
  __syncthreads();
  const int mrow = 8 * (lane >> 4);        // C/D layout: vgpr r -> M = r + 8*(lane/16)
  const int ncol = lane & 15;              //             N = lane % 16
  for (int w = 0; w < 8; ++w) {
    if (wave == w) {
      #pragma unroll
      for (int t = 0; t < 8; ++t)
        #pragma unroll
        for (int r = 0; r < 8; ++r)
          lds[t * 256 + (mrow + r) * 16 + ncol] += C[t][r];
    }
    __syncthreads();
  }
  float* part = ws + PART_OFF + (size_t)blockIdx.x * GRAM_ELEMS;
  for (int e = tid; e < GRAM_ELEMS; e += 256) part[e] = lds[e];
}

__global__ void gram_reduce_kernel(float* __restrict__ ws, int nblk)
{
  int e = blockIdx.x * blockDim.x + threadIdx.x;
  if (e >= GRAM_ELEMS) return;
  float acc = 0.f;
  for (int b = 0; b < nblk; ++b) acc += ws[PART_OFF + (size_t)b * GRAM_ELEMS + e];
  ws[e] = acc;                             // fixed-order -> deterministic
}

__device__ __forceinline__ float gramAt(const float* g, int row, int col)
{
  int i = row >> 4, jp = (col >> 4) - 2, m = row & 15, nn = col & 15;
  return g[(i * 2 + jp) * 256 + m * 16 + nn];
}

__global__ void solve_kernel(const float* __restrict__ ys, const float* __restrict__ theta_p,
                             const float* __restrict__ a_p, float* __restrict__ ws)
{
  if (threadIdx.x != 0 || blockIdx.x != 0) return;
  const float* g = ws;
  const float theta = theta_p[0];
  const float a     = a_p[0];
  float alpha[MHIST], beta[MHIST];

  // loop1 (reverse): alpha_j = (a*(s_j.v) - sum_{k>j} alpha_k * (s_j.y_k)) / ys_j
  for (int j = MHIST - 1; j >= 0; --j) {
    float t = a * gramAt(g, j, 62);                          // s_j . v
    for (int k = j + 1; k < MHIST; ++k) t -= alpha[k] * gramAt(g, j, 32 + k);   // s_j . y_k
    alpha[j] = t / ys[j];
  }
  // loop2: beta_j = ((a*(y_j.v) - sum_k alpha_k*(y_j.y_k))/theta + sum_{k<j}(alpha_k-beta_k)*(s_k.y_j)) / ys_j
  for (int j = 0; j < MHIST; ++j) {
    float t = a * gramAt(g, 32 + j, 62);                     // y_j . v
    for (int k = 0; k < MHIST; ++k) t -= alpha[k] * gramAt(g, 32 + j, 32 + k);  // y_j . y_k
    t /= theta;
    for (int k = 0; k < j; ++k) t += (alpha[k] - beta[k]) * gramAt(g, k, 32 + j); // s_k . y_j
    beta[j] = t / ys[j];
  }
  ws[COEFF_OFF] = a / theta;
  for (int k = 0; k < MHIST; ++k) {
    ws[COEFF_OFF + 1 + k]         = -alpha[k] / theta;       // y_k coefficient
    ws[COEFF_OFF + 1 + MHIST + k] = alpha[k] - beta[k];      // s_k coefficient
  }
}

__global__ __launch_bounds__(256) void combine_kernel(
    const float* __restrict__ v, const float* __restrict__ s, const float* __restrict__ y,
    const float* __restrict__ ws, float* __restrict__ out, long n)
{
  const float cv = ws[COEFF_OFF];
  long i = ((long)blockIdx.x * blockDim.x + threadIdx.x) * 4;
  const long stride = (long)gridDim.x * blockDim.x * 4;
  for (; i + 3 < n; i += stride) {
    float4 av = *(const float4*)(v + i);
    float ax = cv * av.x, ay = cv * av.y, az = cv * av.z, aw = cv * av.w;
    #pragma unroll 3
    for (int k = 0; k < MHIST; ++k) {
      const float cyk = ws[COEFF_OFF + 1 + k];               // uniform -> scalar load
      const float csk = ws[COEFF_OFF + 1 + MHIST + k];
      float4 y4 = *(const float4*)(y + (size_t)k * n + i);
      float4 s4 = *(const float4*)(s + (size_t)k * n + i);
      ax += cyk * y4.x + csk * s4.x;
      ay += cyk * y4.y + csk * s4.y;
      az += cyk * y4.z + csk * s4.z;
      aw += cyk * y4.w + csk * s4.w;
    }
    *(float4*)(out + i) = make_float4(ax, ay, az, aw);
  }
}

extern "C" void kernel_launch(void* const* d_in, const int* in_sizes, int n_in,
                              void* d_out, int out_size, void* d_ws, size_t ws_size,
                              hipStream_t stream)
{
  const float* v     = (const float*)d_in[0];
  const float* s     = (const float*)d_in[1];
  const float* y     = (const float*)d_in[2];
  const float* ys    = (const float*)d_in[3];
  const float* theta = (const float*)d_in[4];
  const float* a     = (const float*)d_in[5];
  float* out = (float*)d_out;
  float* ws  = (float*)d_ws;
  const long n = in_sizes[0];

  zpad_init_kernel<<<1, 64, 0, stream>>>(ws);
  gram_partial_kernel<<<NBLK1, 256, 0, stream>>>(v, s, y, ws, n);
  gram_reduce_kernel<<<(GRAM_ELEMS + 255) / 256, 256, 0, stream>>>(ws, NBLK1);
  solve_kernel<<<1, 32, 0, stream>>>(ys, theta, a, ws);
  long work = n / 4;
  int cgrid = (int)((work + 255) / 256);
  combine_kernel<<<cgrid, 256, 0, stream>>>(v, s, y, ws, out, n);
}